// GATPolicy_6296422056029
// MI455X (gfx1250) — compile-verified
//
#include <hip/hip_runtime.h>
#include <math.h>

#define NEG_SLOPE 0.2f

typedef float v2f __attribute__((ext_vector_type(2)));
typedef float v8f __attribute__((ext_vector_type(8)));

// ---- monotone float<->uint encoding for atomicMax-based segment max ----
__device__ __forceinline__ unsigned fenc(float f) {
    unsigned u = __float_as_uint(f);
    return (u & 0x80000000u) ? ~u : (u | 0x80000000u);
}
__device__ __forceinline__ float fdec(unsigned u) {
    return (u & 0x80000000u) ? __uint_as_float(u ^ 0x80000000u)
                             : __uint_as_float(~u);
}

// edge list with implicit self-loops appended (PyG add_self_loops=True)
__device__ __forceinline__ void edge_sd(const long long* __restrict__ ei, int E,
                                        int e, int& s, int& d) {
    if (e < E) { s = (int)ei[e]; d = (int)ei[E + e]; }
    else       { s = e - E; d = e - E; }
}

// ---------------- Layer 1: h1 = x@W1 ; per-node attention halves ----------------
__global__ void k_node1(const float* __restrict__ x, const float* __restrict__ W1,
                        const float* __restrict__ as1, const float* __restrict__ ad1,
                        float* __restrict__ h1, float* __restrict__ asrc1,
                        float* __restrict__ adst1) {
    __shared__ float ssrc[512];
    __shared__ float sdst[512];
    int n = blockIdx.x;
    int t = threadIdx.x;            // 0..511 = head*128 + ch
    int h = t >> 7, c = t & 127;
    float x0 = x[n * 2 + 0], x1 = x[n * 2 + 1];
    float v = x0 * W1[t] + x1 * W1[512 + t];
    h1[(size_t)n * 512 + t] = v;
    ssrc[t] = v * as1[t];           // att_src1 flat [4,128] == index t
    sdst[t] = v * ad1[t];
    for (int off = 64; off; off >>= 1) {
        __syncthreads();
        if (c < off) { ssrc[t] += ssrc[t + off]; sdst[t] += sdst[t + off]; }
    }
    __syncthreads();
    if (c == 0) { asrc1[n * 4 + h] = ssrc[t]; adst1[n * 4 + h] = sdst[t]; }
}

// ---------------- Layer 1 edge passes ----------------
__global__ void k_e1max(const long long* __restrict__ ei, int E, int E2,
                        const float* __restrict__ asrc1, const float* __restrict__ adst1,
                        float* __restrict__ e1, unsigned* __restrict__ m1) {
    int idx = blockIdx.x * blockDim.x + threadIdx.x;
    if (idx >= E2 * 4) return;
    int e = idx >> 2, h = idx & 3;
    int s, d; edge_sd(ei, E, e, s, d);
    float ev = asrc1[s * 4 + h] + adst1[d * 4 + h];
    ev = ev > 0.f ? ev : NEG_SLOPE * ev;
    e1[idx] = ev;
    atomicMax(&m1[d * 4 + h], fenc(ev));
}

__global__ void k_e1sum(const long long* __restrict__ ei, int E, int E2,
                        float* __restrict__ e1, const unsigned* __restrict__ m1,
                        float* __restrict__ s1) {
    int idx = blockIdx.x * blockDim.x + threadIdx.x;
    if (idx >= E2 * 4) return;
    int e = idx >> 2, h = idx & 3;
    int s, d; edge_sd(ei, E, e, s, d);
    (void)s;
    float p = expf(e1[idx] - fdec(m1[d * 4 + h]));
    e1[idx] = p;
    atomicAdd(&s1[d * 4 + h], p);
}

// one wave32 per (edge, head); 4 channels per lane
__global__ void k_e1aggr(const long long* __restrict__ ei, int E, int E2,
                         const float* __restrict__ e1, const float* __restrict__ s1,
                         const float* __restrict__ h1, float* __restrict__ out1) {
    long long tid = (long long)blockIdx.x * blockDim.x + threadIdx.x;
    int pair = (int)(tid >> 5);
    int lane = (int)(tid & 31);
    if (pair >= E2 * 4) return;
    int e = pair >> 2, h = pair & 3;
    int s, d; edge_sd(ei, E, e, s, d);
    float alpha = e1[pair] / (s1[d * 4 + h] + 1e-16f);
    const float* hs = h1 + (size_t)s * 512 + h * 128;
    float* od = out1 + (size_t)d * 512 + h * 128;
#pragma unroll
    for (int c = 0; c < 128; c += 32)
        atomicAdd(&od[c + lane], hs[c + lane] * alpha);
}

// ---------------- Layer 2 GEMM: h2 = relu(out1 + b1) @ W2  via f32 WMMA ----------------
// [N,512] x [512,128]; one block = 16 rows, 8 waves each own a 16-col slab.
__global__ void k_gemm2(const float* __restrict__ acc, const float* __restrict__ b1,
                        const float* __restrict__ W2, float* __restrict__ h2, int N) {
    __shared__ float sA[16 * 512];   // 32 KB A-tile with bias+relu fused
    int rowTile = blockIdx.x;
    for (int i = threadIdx.x; i < 16 * 512; i += 256) {
        int r = i >> 9, k = i & 511;
        int row = rowTile * 16 + r;
        float av = 0.f;
        if (row < N) {
            av = acc[(size_t)row * 512 + k] + b1[k];
            av = av > 0.f ? av : 0.f;
        }
        sA[i] = av;
    }
    __syncthreads();

    int wave = threadIdx.x >> 5;
    int lane = threadIdx.x & 31;
    int m  = lane & 15;              // A: M row / B,D: N col within tile
    int kh = lane >> 4;              // lane half selects K pair (A/B), M+8 (D)
    int col = wave * 16 + m;

    v8f c = {};
    for (int k0 = 0; k0 < 512; k0 += 4) {
        int ka = k0 + kh * 2;
        v2f a, b;
        a.x = sA[m * 512 + ka];
        a.y = sA[m * 512 + ka + 1];
        b.x = W2[(size_t)ka * 128 + col];
        b.y = W2[(size_t)(ka + 1) * 128 + col];
        c = __builtin_amdgcn_wmma_f32_16x16x4_f32(false, a, false, b,
                                                  (short)0, c, false, false);
    }
#pragma unroll
    for (int v = 0; v < 8; ++v) {
        int row = rowTile * 16 + v + 8 * kh;
        if (row < N) h2[(size_t)row * 128 + col] = c[v];
    }
}

// ---------------- Layer 2: per-node attention halves ----------------
__global__ void k_node2(const float* __restrict__ h2, const float* __restrict__ as2,
                        const float* __restrict__ ad2, float* __restrict__ asrc2,
                        float* __restrict__ adst2) {
    __shared__ float ss[128];
    __shared__ float sd[128];
    int n = blockIdx.x, c = threadIdx.x;
    float v = h2[(size_t)n * 128 + c];
    ss[c] = v * as2[c];
    sd[c] = v * ad2[c];
    for (int off = 64; off; off >>= 1) {
        __syncthreads();
        if (c < off) { ss[c] += ss[c + off]; sd[c] += sd[c + off]; }
    }
    __syncthreads();
    if (c == 0) { asrc2[n] = ss[0]; adst2[n] = sd[0]; }
}

// ---------------- Layer 2 edge passes (H=1) ----------------
__global__ void k_e2max(const long long* __restrict__ ei, int E, int E2,
                        const float* __restrict__ asrc2, const float* __restrict__ adst2,
                        float* __restrict__ e2, unsigned* __restrict__ m2) {
    int e = blockIdx.x * blockDim.x + threadIdx.x;
    if (e >= E2) return;
    int s, d; edge_sd(ei, E, e, s, d);
    float ev = asrc2[s] + adst2[d];
    ev = ev > 0.f ? ev : NEG_SLOPE * ev;
    e2[e] = ev;
    atomicMax(&m2[d], fenc(ev));
}

__global__ void k_e2sum(const long long* __restrict__ ei, int E, int E2,
                        float* __restrict__ e2, const unsigned* __restrict__ m2,
                        float* __restrict__ s2) {
    int e = blockIdx.x * blockDim.x + threadIdx.x;
    if (e >= E2) return;
    int s, d; edge_sd(ei, E, e, s, d);
    (void)s;
    float p = expf(e2[e] - fdec(m2[d]));
    e2[e] = p;
    atomicAdd(&s2[d], p);
}

__global__ void k_e2aggr(const long long* __restrict__ ei, int E, int E2,
                         const float* __restrict__ e2, const float* __restrict__ s2,
                         const float* __restrict__ h2, float* __restrict__ out2) {
    long long tid = (long long)blockIdx.x * blockDim.x + threadIdx.x;
    int e = (int)(tid >> 5);
    int lane = (int)(tid & 31);
    if (e >= E2) return;
    int s, d; edge_sd(ei, E, e, s, d);
    float alpha = e2[e] / (s2[d] + 1e-16f);
    const float* hs = h2 + (size_t)s * 128;
    float* od = out2 + (size_t)d * 128;
#pragma unroll
    for (int c = 0; c < 128; c += 32)
        atomicAdd(&od[c + lane], hs[c + lane] * alpha);
}

// ---------------- Final: logits = relu(out2 + b2) @ Wp + bp ----------------
__global__ void k_final(const float* __restrict__ out2, const float* __restrict__ b2,
                        const float* __restrict__ Wp, const float* __restrict__ bp,
                        float* __restrict__ out) {
    __shared__ float sr[128];
    int n = blockIdx.x, c = threadIdx.x;
    float v = out2[(size_t)n * 128 + c] + b2[c];
    v = v > 0.f ? v : 0.f;
    sr[c] = v * Wp[c];
    for (int off = 64; off; off >>= 1) {
        __syncthreads();
        if (c < off) sr[c] += sr[c + off];
    }
    __syncthreads();
    if (c == 0) out[n] = sr[0] + bp[0];
}

extern "C" void kernel_launch(void* const* d_in, const int* in_sizes, int n_in,
                              void* d_out, int out_size, void* d_ws, size_t ws_size,
                              hipStream_t stream) {
    const float*     x   = (const float*)d_in[0];
    const long long* ei  = (const long long*)d_in[1];   // int64 edge_index [2,E]
    const float*     W1  = (const float*)d_in[2];
    const float*     as1 = (const float*)d_in[3];
    const float*     ad1 = (const float*)d_in[4];
    const float*     b1  = (const float*)d_in[5];
    const float*     W2  = (const float*)d_in[6];
    const float*     as2 = (const float*)d_in[7];
    const float*     ad2 = (const float*)d_in[8];
    const float*     b2  = (const float*)d_in[9];
    const float*     Wp  = (const float*)d_in[10];
    const float*     bp  = (const float*)d_in[11];

    int N  = in_sizes[0] / 2;     // x is [N,2]
    int E  = in_sizes[1] / 2;     // edge_index is [2,E]
    int E2 = E + N;               // + self loops

    // ---- workspace layout (all 4-byte elements); zero-init block contiguous ----
    float* ws = (float*)d_ws;
    size_t o = 0;
    float*    h1    = ws + o; o += (size_t)N * 512;
    float*    zbase = ws + o;
    float*    out1  = ws + o; o += (size_t)N * 512;   // accum, zeroed
    float*    out2  = ws + o; o += (size_t)N * 128;   // accum, zeroed
    unsigned* m1    = (unsigned*)(ws + o); o += (size_t)N * 4;  // enc(-inf)>0
    float*    s1    = ws + o; o += (size_t)N * 4;
    unsigned* m2    = (unsigned*)(ws + o); o += (size_t)N;
    float*    s2    = ws + o; o += (size_t)N;
    size_t zbytes   = (size_t)((ws + o) - zbase) * sizeof(float);
    float*    h2    = ws + o; o += (size_t)N * 128;
    float*    asrc1 = ws + o; o += (size_t)N * 4;
    float*    adst1 = ws + o; o += (size_t)N * 4;
    float*    asrc2 = ws + o; o += (size_t)N;
    float*    adst2 = ws + o; o += (size_t)N;
    float*    e1    = ws + o; o += (size_t)E2 * 4;
    float*    e2    = ws + o; o += (size_t)E2;

    hipMemsetAsync(zbase, 0, zbytes, stream);

    // layer 1
    k_node1<<<N, 512, 0, stream>>>(x, W1, as1, ad1, h1, asrc1, adst1);
    int t1 = E2 * 4;
    k_e1max<<<(t1 + 255) / 256, 256, 0, stream>>>(ei, E, E2, asrc1, adst1, e1, m1);
    k_e1sum<<<(t1 + 255) / 256, 256, 0, stream>>>(ei, E, E2, e1, m1, s1);
    long long th1 = (long long)t1 * 32;
    k_e1aggr<<<(unsigned)((th1 + 255) / 256), 256, 0, stream>>>(ei, E, E2, e1, s1, h1, out1);

    // layer 2 GEMM (f32 WMMA)
    k_gemm2<<<(N + 15) / 16, 256, 0, stream>>>(out1, b1, W2, h2, N);

    // layer 2 attention
    k_node2<<<N, 128, 0, stream>>>(h2, as2, ad2, asrc2, adst2);
    k_e2max<<<(E2 + 255) / 256, 256, 0, stream>>>(ei, E, E2, asrc2, adst2, e2, m2);
    k_e2sum<<<(E2 + 255) / 256, 256, 0, stream>>>(ei, E, E2, e2, m2, s2);
    long long th2 = (long long)E2 * 32;
    k_e2aggr<<<(unsigned)((th2 + 255) / 256), 256, 0, stream>>>(ei, E, E2, e2, s2, h2, out2);

    // readout
    k_final<<<N, 128, 0, stream>>>(out2, b2, Wp, bp, (float*)d_out);
}